// VanillaTopKPooling_47081431499189
// MI455X (gfx1250) — compile-verified
//
#include <hip/hip_runtime.h>
#include <hip/hip_bf16.h>

// Problem constants (from the reference):
//   x: (B=64, N=4096, D=1024) fp32, top-K=16 along N, mean -> out (64, 1024)
#define B_ 64
#define N_ 4096
#define D_ 1024
#define K_ 16
#define UNROLL_ 8

// One thread per (b, d) output. Threads consecutive in d -> each wave32 load
// of row n is 32 consecutive floats = one 128B line (perfect coalescing).
// Data is single-use (1 GiB streamed once): non-temporal loads keep the
// stream out of L2 (TH=NT), plus a speculative global_prefetch_b8 ahead.
//
// Top-16 kept as a sorted-descending register array; insertion uses the
// median-of-3 identity so each (rare, guarded) insert is 15 x v_med3_num_f32
// + 1 x v_max instead of a 45-op compare/cndmask bubble pass.
__global__ __launch_bounds__(256) void topk16_mean_kernel(
    const float* __restrict__ x, float* __restrict__ out) {
  const int tid    = threadIdx.x;
  const int dChunk = blockIdx.x & 3;   // D_/256 = 4 chunks of d per b
  const int b      = blockIdx.x >> 2;
  const int d      = dChunk * 256 + tid;

  const float* p = x + (size_t)b * N_ * D_ + d;

  // m[0] = largest ... m[K-1] = smallest of the current top-16.
  float m[K_];
#pragma unroll
  for (int i = 0; i < K_; ++i) m[i] = -__builtin_inff();

  for (int n = 0; n < N_; n += UNROLL_) {
    // Batch the loads so the wave keeps UNROLL_ lines in flight (s_clause).
    float v[UNROLL_];
#pragma unroll
    for (int u = 0; u < UNROLL_; ++u)
      v[u] = __builtin_nontemporal_load(p + (size_t)(n + u) * D_);

    // Speculative look-ahead prefetch for the next chunk's first line
    // (silently dropped if the translation fails).
    if (n + 2 * UNROLL_ <= N_)
      __builtin_prefetch(p + (size_t)(n + UNROLL_) * D_, 0, 0);

#pragma unroll
    for (int u = 0; u < UNROLL_; ++u) {
      const float val = v[u];
      if (val > m[K_ - 1]) {   // rare path: beats current 16th value
        // Descending j uses the pre-update m[j-1]: exact sorted insert.
#pragma unroll
        for (int j = K_ - 1; j >= 1; --j)
          m[j] = __builtin_amdgcn_fmed3f(m[j - 1], m[j], val);
        m[0] = fmaxf(m[0], val);
      }
    }
  }

  float s = 0.0f;
#pragma unroll
  for (int i = 0; i < K_; ++i) s += m[i];
  out[(size_t)b * D_ + d] = s * (1.0f / (float)K_);
}

extern "C" void kernel_launch(void* const* d_in, const int* in_sizes, int n_in,
                              void* d_out, int out_size, void* d_ws, size_t ws_size,
                              hipStream_t stream) {
  (void)in_sizes; (void)n_in; (void)d_ws; (void)ws_size; (void)out_size;
  const float* x  = (const float*)d_in[0];
  float* out      = (float*)d_out;

  // 64 * (1024/256) = 256 blocks of 256 threads (8 wave32s each) = 2048 waves.
  dim3 grid(B_ * (D_ / 256));
  dim3 block(256);
  topk16_mean_kernel<<<grid, block, 0, stream>>>(x, out);
}